// GHMCLoss_21895743275016
// MI455X (gfx1250) — compile-verified
//
#include <hip/hip_runtime.h>

typedef __attribute__((ext_vector_type(2))) float v2f;
typedef __attribute__((ext_vector_type(8))) float v8f;

#define NBINS 10
#define TPB   256
#define HSTRIDE 11   // 11 float2 slots/thread: dword stride 22, gcd(22,64)=2 -> mild conflicts only

// Per-element GHM update into this thread's private LDS histogram slice.
__device__ __forceinline__ void ghm_elem(float p, float t, float w,
                                         float2* __restrict__ hist, int base) {
  float valid = (w > 0.0f) ? 1.0f : 0.0f;
  float ap  = fabsf(p);
  float e   = expf(-ap);                         // e in (0,1]
  float inv = __builtin_amdgcn_rcpf(1.0f + e);
  float sg  = ((p >= 0.0f) ? 1.0f : e) * inv;    // sigmoid(p), one exp + one rcp
  float g   = fabsf(sg - t);
  int bin = (int)(g * 10.0f);
  bin = bin < 0 ? 0 : (bin > NBINS - 1 ? NBINS - 1 : bin);
  float bce = fmaxf(p, 0.0f) - p * t + log1pf(e); // stable BCE-with-logits
  int idx = base + bin;
  float2 h = hist[idx];                           // ds_load_b64 RMW, no atomics
  h.x += valid * bce;
  h.y += valid;
  hist[idx] = h;
}

__global__ void __launch_bounds__(TPB)
ghm_partial(const float* __restrict__ pred, const float* __restrict__ targ,
            const float* __restrict__ lw, float* __restrict__ wsS,
            unsigned* __restrict__ wsC, long long n) {
  __shared__ float2 hist[TPB * HSTRIDE];          // 22.5 KB of the 320 KB/WGP LDS
  const int tid  = threadIdx.x;
  const int base = tid * HSTRIDE;
  for (int b = 0; b < HSTRIDE; ++b) hist[base + b] = make_float2(0.f, 0.f);

  const long long n4      = n >> 2;
  const long long gstride = (long long)gridDim.x * TPB;
  const long long gid     = (long long)blockIdx.x * TPB + tid;
  const float4* p4 = (const float4*)pred;
  const float4* t4 = (const float4*)targ;
  const float4* w4 = (const float4*)lw;

  for (long long i = gid; i < n4; i += gstride) {
    if (i + gstride < n4) {                       // speculative stream prefetch
      __builtin_prefetch(&p4[i + gstride], 0, 0);
      __builtin_prefetch(&t4[i + gstride], 0, 0);
      __builtin_prefetch(&w4[i + gstride], 0, 0);
    }
    float4 p = p4[i]; float4 t = t4[i]; float4 w = w4[i];   // global_load_b128 x3
    ghm_elem(p.x, t.x, w.x, hist, base);
    ghm_elem(p.y, t.y, w.y, hist, base);
    ghm_elem(p.z, t.z, w.z, hist, base);
    ghm_elem(p.w, t.w, w.w, hist, base);
  }
  for (long long i = (n4 << 2) + gid; i < n; i += gstride)  // scalar tail
    ghm_elem(pred[i], targ[i], lw[i], hist, base);

  // Fold 256 private histograms down to row 0 (log2 tree).
  for (int s = TPB / 2; s > 0; s >>= 1) {
    __syncthreads();
    if (tid < s) {
      for (int b = 0; b < NBINS; ++b) {
        float2 a = hist[tid * HSTRIDE + b];
        float2 c = hist[(tid + s) * HSTRIDE + b];
        a.x += c.x; a.y += c.y;
        hist[tid * HSTRIDE + b] = a;
      }
    }
  }
  __syncthreads();
  if (tid < 16) {                                  // 16-wide rows: bins 10..15 are zero pad
    float s = 0.f; unsigned c = 0;
    if (tid < NBINS) { float2 h = hist[tid]; s = h.x; c = (unsigned)(h.y + 0.5f); }
    wsS[(size_t)blockIdx.x * 16 + tid] = s;
    wsC[(size_t)blockIdx.x * 16 + tid] = c;
  }
}

// Finalize: WMMA f32 16x16x4 reduction of per-block BCE partials (B = ones),
// exact uint sums for counts, then loss = sum_b S_b/counts_b / n_nonempty.
__global__ void __launch_bounds__(32)
ghm_final(const float* __restrict__ wsS, const unsigned* __restrict__ wsC,
          float* __restrict__ out, int nb) {
  __shared__ float    Sb[16];
  __shared__ unsigned Cb[NBINS];
  const int lane = threadIdx.x;
  const int row  = lane & 15;     // A-matrix M index (bin)
  const int kh   = lane >> 4;     // lanes 0-15: K=0,1 ; lanes 16-31: K=2,3

  v8f acc = {0.f, 0.f, 0.f, 0.f, 0.f, 0.f, 0.f, 0.f};
  const v2f bones = {1.0f, 1.0f};                  // B = ones(4x16): layout-agnostic
  for (int i = 0; i < nb; i += 4) {                // D[m][n] += sum_k A[m][k]
    v2f a;
    a.x = wsS[(size_t)(i + 2 * kh + 0) * 16 + row];
    a.y = wsS[(size_t)(i + 2 * kh + 1) * 16 + row];
    acc = __builtin_amdgcn_wmma_f32_16x16x4_f32(false, a, false, bones,
                                                (short)0, acc, false, false);
  }
  // D layout: VGPR r -> M=r (lanes 0-15, N=lane) ; M=8+r (lanes 16-31)
  if (lane == 0)  { for (int r = 0; r < 8; ++r) Sb[r]     = acc[r]; }
  if (lane == 16) { for (int r = 0; r < 8; ++r) Sb[8 + r] = acc[r]; }
  if (lane < NBINS) {
    unsigned c = 0;
    for (int i = 0; i < nb; ++i) c += wsC[(size_t)i * 16 + lane];  // exact integer sum
    Cb[lane] = c;
  }
  __syncthreads();
  if (lane == 0) {
    int nne = 0; float loss = 0.f;
    for (int b = 0; b < NBINS; ++b)
      if (Cb[b] > 0) { ++nne; loss += Sb[b] / (float)Cb[b]; }
    out[0] = (nne > 0) ? loss / (float)nne : 0.f;   // LOSS_WEIGHT = 1.0
  }
}

extern "C" void kernel_launch(void* const* d_in, const int* in_sizes, int n_in,
                              void* d_out, int out_size, void* d_ws, size_t ws_size,
                              hipStream_t stream) {
  (void)n_in; (void)out_size;
  const float* pred = (const float*)d_in[0];
  const float* targ = (const float*)d_in[1];
  const float* lw   = (const float*)d_in[2];
  long long n = (long long)in_sizes[0];

  int nb = 1024;                                   // power of two -> multiple of 4 for K2
  while (nb > 4 && (size_t)nb * 16 * (sizeof(float) + sizeof(unsigned)) > ws_size)
    nb >>= 1;
  float*    wsS = (float*)d_ws;
  unsigned* wsC = (unsigned*)((char*)d_ws + (size_t)nb * 16 * sizeof(float));

  ghm_partial<<<nb, TPB, 0, stream>>>(pred, targ, lw, wsS, wsC, n);
  ghm_final<<<1, 32, 0, stream>>>(wsS, wsC, (float*)d_out, nb);
}